// ScorePredictor_12962211299984
// MI455X (gfx1250) — compile-verified
//
#include <hip/hip_runtime.h>
#include <math.h>

#define D 128

typedef float v2f __attribute__((ext_vector_type(2)));
typedef float v8f __attribute__((ext_vector_type(8)));

// ---------------------------------------------------------------------------
// Kernel 1: H = X @ W   (nrows x 128) = (nrows x 128) x (128 x 128)
// fp32 WMMA path: V_WMMA_F32_16X16X4_F32, wave32.
// Block = 256 threads = 8 waves. Block b owns rows [16b, 16b+16);
// wave w owns cols [16w, 16w+16) (8 waves x 16 cols = 128 cols).
// K-loop: 32 steps of K=4.
//
// Fragment layouts per CDNA5 ISA 7.12.2:
//   A (16x4 f32, 2 VGPRs): lanes 0-15 -> M=lane, {K=0,K=1}; lanes 16-31 -> {K=2,K=3}
//   B (4x16 f32, 2 VGPRs): lanes 0-15 -> N=lane, rows {K=0,K=1}; lanes 16-31 -> {K=2,K=3}
//   C/D (16x16 f32, 8 VGPRs): VGPR v, lane l -> M = v + 8*(l/16), N = l%16
// ---------------------------------------------------------------------------
__global__ __launch_bounds__(256) void xw_gemm_wmma(const float* __restrict__ X,
                                                    const float* __restrict__ W,
                                                    float* __restrict__ H,
                                                    int nrows) {
    const int wave = threadIdx.x >> 5;   // 0..7
    const int lane = threadIdx.x & 31;
    const int half = lane >> 4;          // 0 or 1
    const int l16  = lane & 15;
    const int row0 = blockIdx.x << 4;
    const int col0 = wave << 4;
    if (row0 >= nrows) return;           // block-uniform (EXEC stays all-ones)

    const float* Arow = X + (size_t)(row0 + l16) * D;  // this lane's A row
    const float* Bcol = W + col0 + l16;                // this lane's B column

    v8f c = {};
    #pragma unroll 4
    for (int k = 0; k < D; k += 4) {
        v2f a, b;
        const int ka = k + 2 * half;
        a.x = Arow[ka + 0];
        a.y = Arow[ka + 1];
        b.x = Bcol[(size_t)(ka + 0) * D];
        b.y = Bcol[(size_t)(ka + 1) * D];
        // 8 args: (neg_a, A, neg_b, B, c_mod, C, reuse_a, reuse_b)
        c = __builtin_amdgcn_wmma_f32_16x16x4_f32(false, a, false, b,
                                                  (short)0, c, false, false);
    }

    float* Hout = H + (size_t)(row0 + 8 * half) * D + col0 + l16;
    #pragma unroll
    for (int v = 0; v < 8; ++v)
        Hout[(size_t)v * D] = c[v];
}

// ---------------------------------------------------------------------------
// Kernel 2: per-edge score = sigmoid( dot(H[src], X[dst]) )
// One wave (32 lanes) per edge; lane l handles 4 contiguous floats (float4),
// so each gather row is one coalesced 128B transaction set. H and X both fit
// in the 192MB L2 simultaneously -> gathers hit L2.
// ---------------------------------------------------------------------------
__global__ __launch_bounds__(256) void edge_score(const float* __restrict__ H,
                                                  const float* __restrict__ X,
                                                  const int* __restrict__ src,
                                                  const int* __restrict__ dst,
                                                  float* __restrict__ out,
                                                  int nedges) {
    const int wave = threadIdx.x >> 5;
    const int lane = threadIdx.x & 31;
    const int e = blockIdx.x * 8 + wave;
    if (e >= nedges) return;             // wave-uniform

    const float4* hs = (const float4*)(H + (size_t)src[e] * D);
    const float4* xd = (const float4*)(X + (size_t)dst[e] * D);
    const float4 a = hs[lane];
    const float4 b = xd[lane];
    float s = a.x * b.x + a.y * b.y + a.z * b.z + a.w * b.w;

    // wave32 butterfly reduction
    s += __shfl_xor(s, 16, 32);
    s += __shfl_xor(s, 8, 32);
    s += __shfl_xor(s, 4, 32);
    s += __shfl_xor(s, 2, 32);
    s += __shfl_xor(s, 1, 32);

    if (lane == 0)
        out[e] = 1.0f / (1.0f + __expf(-s));
}

// ---------------------------------------------------------------------------
// Fallback (only if workspace can't hold H = nrows*128 floats): fused VALU
// version, raw = sum_j (sum_i src[i]*W[i][j]) * dst[j]. One wave per edge,
// lane l owns columns {4l..4l+3}.
// ---------------------------------------------------------------------------
__global__ __launch_bounds__(256) void edge_score_fused(const float* __restrict__ X,
                                                        const float* __restrict__ W,
                                                        const int* __restrict__ src,
                                                        const int* __restrict__ dst,
                                                        float* __restrict__ out,
                                                        int nedges) {
    const int wave = threadIdx.x >> 5;
    const int lane = threadIdx.x & 31;
    const int e = blockIdx.x * 8 + wave;
    if (e >= nedges) return;

    const float* sp = X + (size_t)src[e] * D;
    const float* dp = X + (size_t)dst[e] * D;
    const int j = lane * 4;

    float a0 = 0.f, a1 = 0.f, a2 = 0.f, a3 = 0.f;
    for (int i = 0; i < D; ++i) {
        const float si = sp[i];
        const float* wr = W + (size_t)i * D + j;
        a0 += si * wr[0];
        a1 += si * wr[1];
        a2 += si * wr[2];
        a3 += si * wr[3];
    }
    float s = a0 * dp[j] + a1 * dp[j + 1] + a2 * dp[j + 2] + a3 * dp[j + 3];

    s += __shfl_xor(s, 16, 32);
    s += __shfl_xor(s, 8, 32);
    s += __shfl_xor(s, 4, 32);
    s += __shfl_xor(s, 2, 32);
    s += __shfl_xor(s, 1, 32);

    if (lane == 0)
        out[e] = 1.0f / (1.0f + __expf(-s));
}

// ---------------------------------------------------------------------------
// Inputs (setup_inputs order): 0=node_features f32 [N*128], 1=W f32 [128*128],
// 2=src_idx int [E], 3=dst_idx int [E]. Output: f32 [E].
// ---------------------------------------------------------------------------
extern "C" void kernel_launch(void* const* d_in, const int* in_sizes, int n_in,
                              void* d_out, int out_size, void* d_ws, size_t ws_size,
                              hipStream_t stream) {
    const float* X  = (const float*)d_in[0];
    const float* W  = (const float*)d_in[1];
    const int* src  = (const int*)d_in[2];
    const int* dst  = (const int*)d_in[3];
    float* out      = (float*)d_out;

    const int nnodes = in_sizes[0] / D;   // 100000
    const int nedges = in_sizes[2];       // 625000

    const size_t hbytes = (size_t)nnodes * D * sizeof(float);  // 51.2 MB

    if (d_ws != nullptr && ws_size >= hbytes) {
        float* Hbuf = (float*)d_ws;
        const int row_tiles = (nnodes + 15) / 16;              // 6250
        xw_gemm_wmma<<<row_tiles, 256, 0, stream>>>(X, W, Hbuf, nnodes);
        edge_score<<<(nedges + 7) / 8, 256, 0, stream>>>(Hbuf, X, src, dst, out, nedges);
    } else {
        edge_score_fused<<<(nedges + 7) / 8, 256, 0, stream>>>(X, W, src, dst, out, nedges);
    }
}